// MultiHeadUnion_60662118088793
// MI455X (gfx1250) — compile-verified
//
#include <hip/hip_runtime.h>
#include <hip/hip_bf16.h>

// ---------------- problem constants (fixed by reference) ----------------
#define BATCH 4
#define DIM   512
#define NTOK  8192
#define CATC  256
// group 0
#define H0 4
#define F0 32
#define T0 32
#define TD0 3
#define O0 140          // H0*(F0+3)
#define L0 32768        // T0^3
// group 1
#define H1 8
#define F1 16
#define T1 128
#define TD1 2
#define O1 152          // H1*(F1+3)
#define L1 16384        // T1^2

typedef float v2f __attribute__((ext_vector_type(2)));
typedef float v8f __attribute__((ext_vector_type(8)));

__device__ __forceinline__ v8f wmma_f32(v2f a, v2f b, v8f c) {
  // V_WMMA_F32_16X16X4_F32 : D = A(16x4) * B(4x16) + C(16x16), all fp32
  return __builtin_amdgcn_wmma_f32_16x16x4_f32(
      /*neg_a=*/false, a, /*neg_b=*/false, b,
      /*c_mod=*/(short)0, c, /*reuse_a=*/false, /*reuse_b=*/false);
}

// =======================================================================
// Generic GEMM:  C[b, o, n] = sum_k W[o, k] * X[b, k, n]
// wave tile: 32 (M, 2 subtiles) x 64 (N, 4 subtiles) -> 8 WMMA / k-step
// block = 8 waves (2 M-waves x 4 N-waves) = 64M x 256N
// A rows are CLAMPED (not predicated) so the k-loop is branch-free;
// out-of-range rows produce garbage accs that are never stored.
// No software prefetch: __builtin_prefetch can only emit SE/SYS-scope
// global_prefetch_b8, which stops at GL2 where this panel already lives;
// L2->L0 latency is hidden by the ~17-deep outstanding-load pipeline.
// =======================================================================
__global__ __launch_bounds__(256) void gemm_wmma_f32_kernel(
    const float* __restrict__ W, const float* __restrict__ X,
    float* __restrict__ C, int O, int K) {
  const int lane = threadIdx.x & 31;
  const int wave = threadIdx.x >> 5;
  const int wm = wave & 1, wn = wave >> 1;
  const int m0 = blockIdx.x * 64 + wm * 32;
  const int n0 = blockIdx.y * 256 + wn * 64;
  const int b  = blockIdx.z;

  const int khalf = (lane >> 4) * 2;      // 0 or 2
  const int nl    = n0 + (lane & 15);
  const int mr0   = min(m0 + (lane & 15), O - 1);        // clamped A rows
  const int mr1   = min(m0 + 16 + (lane & 15), O - 1);

  const float* wp0 = W + (size_t)mr0 * K + khalf;
  const float* wp1 = W + (size_t)mr1 * K + khalf;
  const float* xp  = X + (size_t)b * K * NTOK + (size_t)khalf * NTOK + nl;

  v8f acc00 = {}, acc01 = {}, acc02 = {}, acc03 = {};
  v8f acc10 = {}, acc11 = {}, acc12 = {}, acc13 = {};

  for (int k = 0; k < K; k += 4) {
    const v2f a0 = *(const v2f*)wp0;
    const v2f a1 = *(const v2f*)wp1;
    v2f b0, b1, b2, b3;
    b0.x = xp[0];    b0.y = xp[NTOK];
    b1.x = xp[16];   b1.y = xp[NTOK + 16];
    b2.x = xp[32];   b2.y = xp[NTOK + 32];
    b3.x = xp[48];   b3.y = xp[NTOK + 48];
    acc00 = wmma_f32(a0, b0, acc00);
    acc01 = wmma_f32(a0, b1, acc01);
    acc02 = wmma_f32(a0, b2, acc02);
    acc03 = wmma_f32(a0, b3, acc03);
    acc10 = wmma_f32(a1, b0, acc10);
    acc11 = wmma_f32(a1, b1, acc11);
    acc12 = wmma_f32(a1, b2, acc12);
    acc13 = wmma_f32(a1, b3, acc13);
    wp0 += 4; wp1 += 4; xp += (size_t)4 * NTOK;
  }

  const int rbase = m0 + (lane >> 4) * 8;
  const int cbase = n0 + (lane & 15);
  float* Cb = C + (size_t)b * O * NTOK;
#pragma unroll
  for (int v = 0; v < 8; ++v) {
    const int r0 = rbase + v;
    if (r0 < O) {
      float* row = Cb + (size_t)r0 * NTOK + cbase;
      row[0] = acc00[v]; row[16] = acc01[v];
      row[32] = acc02[v]; row[48] = acc03[v];
    }
    const int r1 = r0 + 16;
    if (r1 < O) {
      float* row = Cb + (size_t)r1 * NTOK + cbase;
      row[0] = acc10[v]; row[16] = acc11[v];
      row[32] = acc12[v]; row[48] = acc13[v];
    }
  }
}

// =======================================================================
// Per-channel sum / sumsq over (BATCH, C, NTOK); one block per channel.
// =======================================================================
__global__ __launch_bounds__(256) void channel_stats_kernel(
    const float* __restrict__ buf, float* __restrict__ stats, int Cc) {
  const int c = blockIdx.x;
  const int tid = threadIdx.x;
  float s = 0.f, sq = 0.f;
  for (int i = tid; i < BATCH * NTOK; i += 256) {
    const int b = i >> 13;            // / NTOK
    const int n = i & (NTOK - 1);
    const float v = buf[((size_t)b * Cc + c) * NTOK + n];
    s += v; sq += v * v;
  }
  __shared__ float ls[256], lq[256];
  ls[tid] = s; lq[tid] = sq;
  __syncthreads();
  for (int st = 128; st > 0; st >>= 1) {
    if (tid < st) { ls[tid] += ls[tid + st]; lq[tid] += lq[tid + st]; }
    __syncthreads();
  }
  if (tid == 0) { stats[c * 2] = ls[0]; stats[c * 2 + 1] = lq[0]; }
}

__device__ __forceinline__ void bn_coeffs(const float* stats, int ch,
                                          float& mean, float& inv) {
  const float CNT = (float)(BATCH * NTOK);
  mean = stats[ch * 2] / CNT;
  const float var = stats[ch * 2 + 1] / CNT - mean * mean;
  inv = rsqrtf(var + 1e-5f);
}

// =======================================================================
// Keys: BN(keys) + pcd, tW projection, tanh lattice -> (w, cell)
// =======================================================================
template <int TD, int T, int H>
__global__ __launch_bounds__(256) void keys_kernel(
    const float* __restrict__ kv, const float* __restrict__ stats,
    const float* __restrict__ pcd,
    const float* __restrict__ kbg, const float* __restrict__ kbb,
    const float* __restrict__ tW, const float* __restrict__ tb,
    float* __restrict__ wout, int* __restrict__ cellout, int O) {
  const int gid = blockIdx.x * 256 + threadIdx.x;
  const int n = gid & (NTOK - 1);
  const int bh = gid >> 13;
  const int h = bh % H, b = bh / H;

  float kk[3];
#pragma unroll
  for (int c = 0; c < 3; ++c) {
    const int ch = h * 3 + c;
    const float raw = kv[((size_t)b * O + ch) * NTOK + n];
    float mean, inv; bn_coeffs(stats, ch, mean, inv);
    const float kr = (raw - mean) * inv * kbg[ch] + kbb[ch];
    kk[c] = pcd[((size_t)b * 3 + c) * NTOK + n] + kr;
  }

  float wprod = 1.0f;
  int cell = 0;
#pragma unroll
  for (int t = 0; t < TD; ++t) {
    float kt = tb[h * TD + t];
#pragma unroll
    for (int c = 0; c < 3; ++c) kt += tW[(h * TD + t) * 3 + c] * kk[c];
    const float lat = tanhf(kt);
    const float pos = (lat + 1.0f) * 0.5f * (float)(T - 1);
    float idx = rintf(pos);                      // round-half-even == jnp.round
    idx = fminf(fmaxf(idx, 0.0f), (float)(T - 1));
    const float lc = pos - idx;
    wprod *= (1.0f - fabsf(lc));
    cell = cell * T + (int)idx;
  }
  wout[gid] = wprod;
  cellout[gid] = cell;
}

// =======================================================================
// Scatter: z[b,h,f,cell] += BN(values)[b,h,f,n] * w[b,h,n]
// z layout (B, H, F, L) so conv B-fragment loads are unit-stride in cell.
// =======================================================================
template <int F, int H>
__global__ __launch_bounds__(256) void scatter_kernel(
    const float* __restrict__ kv, const float* __restrict__ stats,
    const float* __restrict__ vbg, const float* __restrict__ vbb,
    const float* __restrict__ wbuf, const int* __restrict__ cellbuf,
    float* __restrict__ z, int O, int L) {
  const int gid = blockIdx.x * 256 + threadIdx.x;
  const int n = gid & (NTOK - 1);
  const int bh = gid >> 13;
  const int h = bh % H, b = bh / H;
  const float wv = wbuf[gid];
  const int cell = cellbuf[gid];
  float* zb = z + ((size_t)(b * H + h) * F) * L + cell;
#pragma unroll 4
  for (int f = 0; f < F; ++f) {
    const int ch = h * F + f;
    const int kvch = H * 3 + ch;
    const float raw = kv[((size_t)b * O + kvch) * NTOK + n];
    float mean, inv; bn_coeffs(stats, kvch, mean, inv);
    const float val = (raw - mean) * inv * vbg[ch] + vbb[ch];
    unsafeAtomicAdd(zb + (size_t)f * L, val * wv);   // global_atomic_add_f32
  }
}

// =======================================================================
// Grouped conv (pad=1) as implicit GEMM over taps with fp32 WMMA.
// per wave: 16 out-channels x 16 cells; K = F per tap.
// Weight tile (16 oc x F ic x NTAP) staged in LDS, shared by all 8 waves.
// Zero-padding handled by clamped address + multiplicative mask (branch-free).
// =======================================================================
template <int TD, int T, int F, int H>
__global__ __launch_bounds__(256) void conv_wmma_kernel(
    const float* __restrict__ z, const float* __restrict__ cw,
    const float* __restrict__ cb, float* __restrict__ zc) {
  constexpr int L = (TD == 3) ? T * T * T : T * T;
  constexpr int NTAP = (TD == 3) ? 27 : 9;
  constexpr int MT = F / 16;

  __shared__ float lw[16 * F * NTAP];    // [tap][ic][ocl]

  const int lane = threadIdx.x & 31;
  const int wave = threadIdx.x >> 5;
  const int hf = blockIdx.y;
  const int h = hf / MT;
  const int oc0 = (hf % MT) * 16;
  const int b = blockIdx.z;

  // -------- cooperative weight stage: lw[(tap*F+ic)*16 + ocl] --------
  for (int e = threadIdx.x; e < 16 * F * NTAP; e += 256) {
    const int ocl = e & 15;
    const int rest = e >> 4;
    const int ic = rest % F;
    const int tap = rest / F;
    lw[e] = cw[((size_t)(h * F + oc0 + ocl) * F + ic) * NTAP + tap];
  }
  __syncthreads();

  const int cell0 = (blockIdx.x * 8 + wave) * 16;
  const int cx0 = cell0 % T;
  const int cy0 = (cell0 / T) % T;
  const int cz0 = (TD == 3) ? (cell0 / (T * T)) : 0;
  const int lx = cx0 + (lane & 15);
  const int khalf = (lane >> 4) * 2;

  const float* zb = z + (size_t)(b * H + h) * F * L;
  const float* lwa = lw + (lane & 15);

  v8f acc = {};
  for (int tap = 0; tap < NTAP; ++tap) {
    int dz, dy, dx;
    if (TD == 3) { dz = tap / 9; dy = (tap / 3) % 3; dx = tap % 3; }
    else         { dz = 0;       dy = tap / 3;       dx = tap % 3; }
    const int nx = lx + dx - 1;
    const int ny = cy0 + dy - 1;
    const int nz = (TD == 3) ? (cz0 + dz - 1) : 0;
    const bool rowok = (ny >= 0) && (ny < T) &&
                       (TD == 2 || (nz >= 0 && nz < T));
    const bool ok = rowok && (nx >= 0) && (nx < T);
    const float maskf = ok ? 1.0f : 0.0f;
    const int nxc = min(max(nx, 0), T - 1);   // clamped: always a legal addr
    const int rowbase = rowok ? (((TD == 3) ? (nz * T + ny) : ny) * T) : 0;
    const int cellix = rowbase + nxc;
    const float* lwt = lwa + tap * F * 16;
#pragma unroll
    for (int ic = 0; ic < F; ic += 4) {
      const int ka = ic + khalf;
      v2f a;
      a.x = lwt[ka * 16];             // ds_load_b32
      a.y = lwt[(ka + 1) * 16];
      v2f bb;
      bb.x = zb[(size_t)ka * L + cellix] * maskf;
      bb.y = zb[(size_t)(ka + 1) * L + cellix] * maskf;
      acc = wmma_f32(a, bb, acc);
    }
  }

  const int cstore = cell0 + (lane & 15);
#pragma unroll
  for (int v = 0; v < 8; ++v) {
    const int oc = oc0 + v + (lane >> 4) * 8;
    zc[((size_t)(b * H + h) * F + oc) * L + cstore] = acc[v] + cb[h * F + oc];
  }
}

// =======================================================================
// Gather: cat[b, base + h*F + f, n] = zc[b,h,f,cell] * w   (pre-BN)
// =======================================================================
template <int F, int H>
__global__ __launch_bounds__(256) void gather_kernel(
    const float* __restrict__ zc, const float* __restrict__ wbuf,
    const int* __restrict__ cellbuf, float* __restrict__ cat,
    int L, int chanBase) {
  const int gid = blockIdx.x * 256 + threadIdx.x;
  const int n = gid & (NTOK - 1);
  const int bh = gid >> 13;
  const int h = bh % H, b = bh / H;
  const float wv = wbuf[gid];
  const int cell = cellbuf[gid];
  const float* zcb = zc + (size_t)(b * H + h) * F * L + cell;
#pragma unroll 4
  for (int f = 0; f < F; ++f) {
    const float g = zcb[(size_t)f * L];
    cat[((size_t)b * CATC + chanBase + h * F + f) * NTOK + n] = g * wv;
  }
}

// BN + ReLU in place on channels [chanBase, chanBase+nChan) of (B, Cc, N)
__global__ __launch_bounds__(256) void bn_relu_kernel(
    float* __restrict__ buf, const float* __restrict__ stats,
    const float* __restrict__ g, const float* __restrict__ bb,
    int Cc, int chanBase, int nChan) {
  const size_t gid = (size_t)blockIdx.x * 256 + threadIdx.x;
  const int n = (int)(gid & (NTOK - 1));
  const int bc = (int)(gid >> 13);
  const int c = bc % nChan, b = bc / nChan;
  const int ch = chanBase + c;
  float mean, inv; bn_coeffs(stats, ch, mean, inv);
  const size_t idx = ((size_t)b * Cc + ch) * NTOK + n;
  const float v = (buf[idx] - mean) * inv * g[c] + bb[c];
  buf[idx] = fmaxf(v, 0.0f);
}

// out = x + relu(BN(out))  (in-place on d_out, which holds the after-GEMM)
__global__ __launch_bounds__(256) void final_residual_kernel(
    float* __restrict__ out, const float* __restrict__ x,
    const float* __restrict__ stats, const float* __restrict__ g,
    const float* __restrict__ bb) {
  const size_t gid = (size_t)blockIdx.x * 256 + threadIdx.x;
  const int c = (int)((gid >> 13) & (DIM - 1));
  float mean, inv; bn_coeffs(stats, c, mean, inv);
  const float v = (out[gid] - mean) * inv * g[c] + bb[c];
  out[gid] = x[gid] + fmaxf(v, 0.0f);
}

// =======================================================================
extern "C" void kernel_launch(void* const* d_in, const int* in_sizes, int n_in,
                              void* d_out, int out_size, void* d_ws, size_t ws_size,
                              hipStream_t stream) {
  const float* x    = (const float*)d_in[0];
  const float* pcd  = (const float*)d_in[1];
  const float* kv_w0 = (const float*)d_in[2];
  const float* kbg0 = (const float*)d_in[3];
  const float* kbb0 = (const float*)d_in[4];
  const float* vbg0 = (const float*)d_in[5];
  const float* vbb0 = (const float*)d_in[6];
  const float* tW0  = (const float*)d_in[7];
  const float* tb0  = (const float*)d_in[8];
  const float* cw0  = (const float*)d_in[9];
  const float* cb0  = (const float*)d_in[10];
  const float* ag0  = (const float*)d_in[11];
  const float* ab0  = (const float*)d_in[12];
  const float* kv_w1 = (const float*)d_in[13];
  const float* kbg1 = (const float*)d_in[14];
  const float* kbb1 = (const float*)d_in[15];
  const float* vbg1 = (const float*)d_in[16];
  const float* vbb1 = (const float*)d_in[17];
  const float* tW1  = (const float*)d_in[18];
  const float* tb1  = (const float*)d_in[19];
  const float* cw1  = (const float*)d_in[20];
  const float* cb1  = (const float*)d_in[21];
  const float* ag1  = (const float*)d_in[22];
  const float* ab1  = (const float*)d_in[23];
  const float* after_w = (const float*)d_in[24];
  const float* abn_g = (const float*)d_in[25];
  const float* abn_b = (const float*)d_in[26];
  float* out = (float*)d_out;

  // ---- workspace layout (floats), buffers reused across the two groups ----
  float* ws = (float*)d_ws;
  float* kv      = ws;                                    // 4,980,736 (O1*B*N)
  float* wbuf    = kv + (size_t)O1 * BATCH * NTOK;        //   262,144
  int*   cellbuf = (int*)(wbuf + (size_t)BATCH * H1 * NTOK);
  float* stats   = (float*)(cellbuf + (size_t)BATCH * H1 * NTOK); // 1,024
  float* zbuf    = stats + 1024;                          // 16,777,216
  float* zcbuf   = zbuf + (size_t)BATCH * H0 * F0 * L0;   // 16,777,216
  float* catbuf  = zcbuf + (size_t)BATCH * H0 * F0 * L0;  //  8,388,608

  // ---------------- group 0 (H=4, F=32, T=32, 3D) ----------------
  gemm_wmma_f32_kernel<<<dim3(3, NTOK / 256, BATCH), 256, 0, stream>>>(
      kv_w0, x, kv, O0, DIM);
  channel_stats_kernel<<<O0, 256, 0, stream>>>(kv, stats, O0);
  keys_kernel<TD0, T0, H0><<<(BATCH * H0 * NTOK) / 256, 256, 0, stream>>>(
      kv, stats, pcd, kbg0, kbb0, tW0, tb0, wbuf, cellbuf, O0);
  hipMemsetAsync(zbuf, 0, (size_t)BATCH * H0 * F0 * L0 * sizeof(float), stream);
  scatter_kernel<F0, H0><<<(BATCH * H0 * NTOK) / 256, 256, 0, stream>>>(
      kv, stats, vbg0, vbb0, wbuf, cellbuf, zbuf, O0, L0);
  conv_wmma_kernel<TD0, T0, F0, H0>
      <<<dim3(L0 / 128, (F0 / 16) * H0, BATCH), 256, 0, stream>>>(
          zbuf, cw0, cb0, zcbuf);
  gather_kernel<F0, H0><<<(BATCH * H0 * NTOK) / 256, 256, 0, stream>>>(
      zcbuf, wbuf, cellbuf, catbuf, L0, 0);

  // ---------------- group 1 (H=8, F=16, T=128, 2D) ----------------
  gemm_wmma_f32_kernel<<<dim3(3, NTOK / 256, BATCH), 256, 0, stream>>>(
      kv_w1, x, kv, O1, DIM);
  channel_stats_kernel<<<O1, 256, 0, stream>>>(kv, stats, O1);
  keys_kernel<TD1, T1, H1><<<(BATCH * H1 * NTOK) / 256, 256, 0, stream>>>(
      kv, stats, pcd, kbg1, kbb1, tW1, tb1, wbuf, cellbuf, O1);
  hipMemsetAsync(zbuf, 0, (size_t)BATCH * H1 * F1 * L1 * sizeof(float), stream);
  scatter_kernel<F1, H1><<<(BATCH * H1 * NTOK) / 256, 256, 0, stream>>>(
      kv, stats, vbg1, vbb1, wbuf, cellbuf, zbuf, O1, L1);
  conv_wmma_kernel<TD1, T1, F1, H1>
      <<<dim3(L1 / 128, (F1 / 16) * H1, BATCH), 256, 0, stream>>>(
          zbuf, cw1, cb1, zcbuf);
  gather_kernel<F1, H1><<<(BATCH * H1 * NTOK) / 256, 256, 0, stream>>>(
      zcbuf, wbuf, cellbuf, catbuf, L1, H0 * F0);

  // ---------------- BN+ReLU on both halves of cat ----------------
  channel_stats_kernel<<<CATC, 256, 0, stream>>>(catbuf, stats, CATC);
  bn_relu_kernel<<<(BATCH * 128 * NTOK) / 256, 256, 0, stream>>>(
      catbuf, stats, ag0, ab0, CATC, 0, 128);
  bn_relu_kernel<<<(BATCH * 128 * NTOK) / 256, 256, 0, stream>>>(
      catbuf, stats, ag1, ab1, CATC, 128, 128);

  // ---------------- after GEMM (into d_out) + BN + ReLU + residual ------
  gemm_wmma_f32_kernel<<<dim3(DIM / 64, NTOK / 256, BATCH), 256, 0, stream>>>(
      after_w, catbuf, out, DIM, CATC);
  channel_stats_kernel<<<DIM, 256, 0, stream>>>(out, stats, DIM);
  final_residual_kernel<<<(BATCH * DIM * NTOK) / 256, 256, 0, stream>>>(
      out, x, stats, abn_g, abn_b);
}